// MLPAttention_27650999451864
// MI455X (gfx1250) — compile-verified
//
#include <hip/hip_runtime.h>
#include <hip/hip_bf16.h>
#include <math.h>

typedef __attribute__((ext_vector_type(2))) float v2f;
typedef __attribute__((ext_vector_type(4))) float v4f;
typedef __attribute__((ext_vector_type(8))) float v8f;
typedef __attribute__((ext_vector_type(4))) unsigned int u32x4;
typedef __attribute__((ext_vector_type(8))) unsigned int u32x8;

#define T_DIM 64
#define B_DIM 16
#define S_DIM 256
#define D_DIM 512

__device__ __forceinline__ float fast_tanh(float x) {
#if __has_builtin(__builtin_amdgcn_tanhf)
  return __builtin_amdgcn_tanhf(x);      // v_tanh_f32 (gfx1250 TRANS op)
#elif __has_builtin(__builtin_amdgcn_tanh_f32)
  return __builtin_amdgcn_tanh_f32(x);
#else
  return tanhf(x);
#endif
}

// Issue a 2D TDM tile load: rows x cols f32 tile (row-major, stride = cols)
// from `gaddr` into the LDS array at `lds_base`, with 1-DWORD LDS padding
// inserted after every 256 DWORDs (pad_interval code 7, pad_amount code 0) ->
// each 256-float chunk occupies 257 floats in LDS (conflict-free column reads).
// D# layout per CDNA5 ISA; groups 2/3 omitted (2D).
//
// NOTE: lds_addr is derived from the array pointer itself (low 32 bits of the
// generic address == wave-relative LDS byte offset, ISA §10.2). This ptrtoint
// also ESCAPES the LDS array into the asm operand, so alias analysis must
// assume the asm (with "memory" clobber) writes it — without this, loads from
// the never-explicitly-stored LDS tile get folded to undef (seen in round 2).
__device__ __forceinline__ void tdm_load_tile_f32(unsigned long long gaddr,
                                                  const float* lds_base,
                                                  unsigned int rows,
                                                  unsigned int cols) {
  const unsigned int lds_off = (unsigned int)(uintptr_t)lds_base;
  u32x4 g0;
  g0[0] = 1u;                                   // count=1, user descriptor
  g0[1] = lds_off;                              // lds_addr (bytes)
  g0[2] = (unsigned int)gaddr;                  // global_addr[31:0]
  g0[3] = (unsigned int)(gaddr >> 32) | (2u << 30);  // global_addr[56:32] | type=2
  u32x8 g1;
  g1[0] = (2u << 16)        // data_size = 4 bytes
        | (1u << 20)        // pad_enable
        | (7u << 22)        // pad_interval code 7 = 256 DWORDs
        | (0u << 25);       // pad_amount code 0 = 1 DWORD
  g1[1] = (cols & 0xFFFFu) << 16;               // tensor_dim0 [63:48]
  g1[2] = (cols >> 16) | ((rows & 0xFFFFu) << 16);   // dim0 hi | tensor_dim1 lo
  g1[3] = (rows >> 16) | (cols << 16);          // dim1 hi | tile_dim0
  g1[4] = rows;                                 // tile_dim1 | tile_dim2=0
  g1[5] = cols;                                 // tensor_dim0_stride [31:0]
  g1[6] = 0u;
  g1[7] = 0u;
  asm volatile("tensor_load_to_lds %0, %1" :: "s"(g0), "s"(g1) : "memory");
}

// Padded-LDS indexing: element (r, c) of a rows x cols f32 tile loaded by
// tdm_load_tile_f32 (1-DWORD pad per 256-DWORD chunk).
#define TDM_IDX(r, c, cols) (((r) * ((cols) >> 8) + ((c) >> 8)) * 257 + ((c) & 255))

// -------------------------------------------------------------------------
// Kernel 1: h[m, n] = input[m, :] @ W[:, n] + bias[n]
// M = T*B = 1024 (row m = t*B + b), N = K = 512. f32 WMMA 16x16x4.
// A-tile (16x512) DMA'd to LDS by the Tensor Data Mover (wave 0 issues,
// s_wait_tensorcnt, workgroup barrier). h stored transposed to (B, T, D).
// -------------------------------------------------------------------------
__global__ __launch_bounds__(256) void combine_gemm(
    const float* __restrict__ input, const float* __restrict__ W,
    const float* __restrict__ bias, float* __restrict__ h) {
  __shared__ float As[16 * 2 * 257];   // 16 rows x 512, TDM-padded
  const int tid  = threadIdx.x;
  const int lane = tid & 31;
  const int wave = tid >> 5;
  const int mtile  = blockIdx.x >> 2;  // 0..63
  const int ngroup = blockIdx.x & 3;   // 0..3  (8 n-tiles per block, 1 per wave)

  if (wave == 0) {
    unsigned long long ga =
        (unsigned long long)(uintptr_t)(input + (size_t)(mtile * 16) * D_DIM);
    tdm_load_tile_f32(ga, As, 16u, (unsigned int)D_DIM);
    __builtin_amdgcn_s_wait_tensorcnt(0);
  }
  __syncthreads();

  const int ncol0 = (ngroup * 8 + wave) * 16;
  const int mrow  = lane & 15;
  const int koff  = (lane >> 4) << 1;  // lanes 0-15 -> K{0,1}; 16-31 -> K{2,3}
  v8f acc = {};
  for (int k0 = 0; k0 < D_DIM; k0 += 4) {
    const int c = k0 + koff;           // even; c and c+1 share a 256-chunk
    const int ai = TDM_IDX(mrow, c, D_DIM);
    v2f a, bm;
    a.x  = As[ai];
    a.y  = As[ai + 1];
    bm.x = W[(size_t)c * D_DIM + ncol0 + mrow];
    bm.y = W[(size_t)(c + 1) * D_DIM + ncol0 + mrow];
    acc = __builtin_amdgcn_wmma_f32_16x16x4_f32(false, a, false, bm,
                                                (short)0, acc, false, false);
  }

  const int ncol  = ncol0 + mrow;
  const float bc  = bias[ncol];
  const int mbase = mtile * 16 + ((lane >> 4) << 3);  // +8 for upper half-wave
#pragma unroll
  for (int i = 0; i < 8; ++i) {
    int m = mbase + i;
    int t = m >> 4, bb = m & 15;                      // B == 16
    h[((size_t)(bb * T_DIM + t)) * D_DIM + ncol] = acc[i] + bc;
  }
}

// -------------------------------------------------------------------------
// Kernel 2: logits[b,t,s] = sum_d v_w[d]*tanh(h[b,t,d]+context[b,s,d]) + v_b,
// then softmax over s. One block per (b,t); 8 waves each own s = wave+8k.
// h-row / v_w register-resident via b128 loads (lane covers d = 4*lane+128j+k);
// context streamed as b128 with global_prefetch of the wave's next row.
// This stage dominates: 1.3e8 v_tanh_f32 + FMAs (TRANS-bound, not memory).
// -------------------------------------------------------------------------
__global__ __launch_bounds__(256) void score_softmax(
    const float* __restrict__ h, const float* __restrict__ context,
    const float* __restrict__ v_w, const float* __restrict__ v_b_p,
    float* __restrict__ attn) {
  __shared__ float logits[S_DIM];
  __shared__ float red[256];
  const int tid  = threadIdx.x;
  const int lane = tid & 31;
  const int wave = tid >> 5;
  const int bx   = blockIdx.x;      // b*T + t
  const int b    = bx >> 6;
  const int t    = bx & 63;
  const float v_b = v_b_p[0];

  const float* hrow = h + ((size_t)(b * T_DIM + t)) * D_DIM;
  v4f hv[4], vw[4];
#pragma unroll
  for (int j = 0; j < 4; ++j) {
    hv[j] = *(const v4f*)(hrow + 128 * j + 4 * lane);
    vw[j] = *(const v4f*)(v_w + 128 * j + 4 * lane);
  }

  for (int s = wave; s < S_DIM; s += 8) {
    const float* crow = context + ((size_t)(b * S_DIM + s)) * D_DIM;
    if (s + 8 < S_DIM)                               // wave-uniform condition
      __builtin_prefetch(crow + 8 * D_DIM + 4 * lane, 0, 1);  // global_prefetch_b8
    float acc = 0.f;
#pragma unroll
    for (int j = 0; j < 4; ++j) {
      v4f c4 = *(const v4f*)(crow + 128 * j + 4 * lane);
      acc = fmaf(vw[j].x, fast_tanh(hv[j].x + c4.x), acc);
      acc = fmaf(vw[j].y, fast_tanh(hv[j].y + c4.y), acc);
      acc = fmaf(vw[j].z, fast_tanh(hv[j].z + c4.z), acc);
      acc = fmaf(vw[j].w, fast_tanh(hv[j].w + c4.w), acc);
    }
#pragma unroll
    for (int off = 16; off > 0; off >>= 1)
      acc += __shfl_xor(acc, off, 32);
    if (lane == 0) logits[s] = acc + v_b;
  }
  __syncthreads();

  // softmax over 256 logits (one element per thread)
  red[tid] = logits[tid];
  __syncthreads();
  for (int st = 128; st > 0; st >>= 1) {
    if (tid < st) red[tid] = fmaxf(red[tid], red[tid + st]);
    __syncthreads();
  }
  const float mx = red[0];
  __syncthreads();
  const float e = __expf(logits[tid] - mx);
  red[tid] = e;
  __syncthreads();
  for (int st = 128; st > 0; st >>= 1) {
    if (tid < st) red[tid] += red[tid + st];
    __syncthreads();
  }
  const float inv = 1.0f / red[0];
  attn[(size_t)bx * S_DIM + tid] = e * inv;
}

// -------------------------------------------------------------------------
// Kernel 3: weighted[b, t, :] = attn[b, t, :] @ values[b, :, :]
// Per-batch GEMM M=64, K=256, N=512 on f32 WMMA 16x16x4.
// attn tile (16x256) DMA'd to LDS by the TDM with pad -> stride-257 rows
// (identical to the manual +1 padding). Output written (T,B,D).
// -------------------------------------------------------------------------
__global__ __launch_bounds__(256) void weighted_gemm(
    const float* __restrict__ attn, const float* __restrict__ values,
    float* __restrict__ out) {
  __shared__ float At[16 * 257];       // 16 rows x 256, TDM-padded (stride 257)
  const int tid  = threadIdx.x;
  const int lane = tid & 31;
  const int wave = tid >> 5;
  const int bx   = blockIdx.x;       // 256 blocks: b(16) x ttile(4) x ngroup(4)
  const int b      = bx >> 4;
  const int ttile  = (bx >> 2) & 3;
  const int ngroup = bx & 3;

  if (wave == 0) {
    unsigned long long ga = (unsigned long long)(uintptr_t)(
        attn + ((size_t)(b * T_DIM + ttile * 16)) * S_DIM);
    tdm_load_tile_f32(ga, At, 16u, (unsigned int)S_DIM);
    __builtin_amdgcn_s_wait_tensorcnt(0);
  }
  __syncthreads();

  const int ncol0 = (ngroup * 8 + wave) * 16;
  const int mrow  = lane & 15;
  const int koff  = (lane >> 4) << 1;
  v8f acc = {};
  for (int k0 = 0; k0 < S_DIM; k0 += 4) {
    const int c  = k0 + koff;
    const int ai = mrow * 257 + c;     // TDM_IDX(mrow, c, 256)
    v2f a, bm;
    a.x  = At[ai];
    a.y  = At[ai + 1];
    bm.x = values[((size_t)(b * S_DIM + c)) * D_DIM + ncol0 + mrow];
    bm.y = values[((size_t)(b * S_DIM + c + 1)) * D_DIM + ncol0 + mrow];
    acc = __builtin_amdgcn_wmma_f32_16x16x4_f32(false, a, false, bm,
                                                (short)0, acc, false, false);
  }

  const int ncol  = ncol0 + mrow;
  const int tbase = ttile * 16 + ((lane >> 4) << 3);
#pragma unroll
  for (int i = 0; i < 8; ++i) {
    int t = tbase + i;
    out[((size_t)(t * B_DIM + b)) * D_DIM + ncol] = acc[i];
  }
}

// -------------------------------------------------------------------------
extern "C" void kernel_launch(void* const* d_in, const int* in_sizes, int n_in,
                              void* d_out, int out_size, void* d_ws, size_t ws_size,
                              hipStream_t stream) {
  (void)in_sizes; (void)n_in; (void)out_size;
  const float* input   = (const float*)d_in[0];
  const float* context = (const float*)d_in[1];
  const float* values  = (const float*)d_in[2];
  const float* W_comb  = (const float*)d_in[3];
  const float* b_comb  = (const float*)d_in[4];
  const float* v_w     = (const float*)d_in[5];
  const float* v_b     = (const float*)d_in[6];

  float* out          = (float*)d_out;
  float* out_weighted = out;                                   // (T,B,D)
  float* out_attn     = out + (size_t)T_DIM * B_DIM * D_DIM;   // (B*T, S)

  // h scratch: (B, T, D) f32 = 2 MB. Prefer d_ws; otherwise reuse the
  // weighted-output region (dead until weighted_gemm fully overwrites it).
  const size_t h_bytes = (size_t)B_DIM * T_DIM * D_DIM * sizeof(float);
  float* h = (ws_size >= h_bytes) ? (float*)d_ws : out_weighted;

  combine_gemm <<<256,  256, 0, stream>>>(input, W_comb, b_comb, h);
  score_softmax<<<1024, 256, 0, stream>>>(h, context, v_w, v_b, out_attn);
  weighted_gemm<<<256,  256, 0, stream>>>(out_attn, values, out_weighted);
}